// HyperAttn_37726992728298
// MI455X (gfx1250) — compile-verified
//
#include <hip/hip_runtime.h>
#include <cstdint>

// ---- problem constants (match setup_inputs: B=16, N=32768, D=64) ----
#define CHUNKS        128     // chunks per batch -> 2048 workgroups total
#define PSTRIDE       68      // floats per partial: 4 scalars (m,s,den,pad) + 64 num
#define WAVES         4       // 128-thread block = 4 wave32
#define ROWS_PER_WAVE 8       // key/value rows per wave per tile (one pass)
#define TILE_ROWS     (WAVES * ROWS_PER_WAVE)   // 32 rows per block tile
#define DEPTH         3       // async pipeline depth (prefetch 2 tiles ahead)

// ---- CDNA5 async global->LDS DMA (ASYNCcnt-tracked), inline asm ----
__device__ __forceinline__ void async_b128(uint32_t ldsOff, const void* g) {
  // dst = LDS byte offset (VGPR), src = 64-bit global address (VGPR pair)
  // NT hint: 256MB streamed once > 192MB L2 -> don't pollute.
  asm volatile("global_load_async_to_lds_b128 %0, %1, off th:TH_LOAD_NT"
               :: "v"(ldsOff), "v"(g) : "memory");
}
__device__ __forceinline__ void wait_async_le16() {
  asm volatile("s_wait_asynccnt 0x10" ::: "memory");
}
__device__ __forceinline__ void wait_async_le8() {
  asm volatile("s_wait_asynccnt 0x8" ::: "memory");
}
__device__ __forceinline__ void wait_async_0() {
  asm volatile("s_wait_asynccnt 0x0" ::: "memory");
}

// Pass 1: per (batch, chunk) streaming online-softmax partial reduction.
__global__ void __launch_bounds__(128)
hyper_attn_partial(const float* __restrict__ key,
                   const float* __restrict__ query,
                   const float* __restrict__ value,
                   const float* __restrict__ beta_p,
                   const float* __restrict__ c_p,
                   float* __restrict__ partials,
                   int N)
{
  // 16B segments within each 256B row are rotate-swizzled by row -> uniform
  // 2x banking for the b128 read phase (optimal: 512B/instr = 2 bank sweeps).
  __shared__ __align__(16) float sK[DEPTH][WAVES][ROWS_PER_WAVE][64]; // 24 KB
  __shared__ __align__(16) float sV[DEPTH][WAVES][ROWS_PER_WAVE][64]; // 24 KB
  __shared__ float sM[128 * 20];                                      // 10 KB

  const int tid  = threadIdx.x;
  const int lane = tid & 31;
  const int wave = tid >> 5;
  const int quad = lane >> 2;   // row-within-pass (0..7)
  const int qs   = lane & 3;    // dim group: owns dims [qs*16, qs*16+16)

  const int b      = blockIdx.x / CHUNKS;
  const int chunk  = blockIdx.x % CHUNKS;
  const int nChunk = N / CHUNKS;              // 256
  const int tiles  = nChunk / TILE_ROWS;      // 8
  const int rowBase0 = chunk * nChunk;

  const float* keyB = key   + (size_t)b * N * 64;
  const float* valB = value + (size_t)b * N * 64;

  const float beta = beta_p[0];
  const float cc   = c_p[0];

  // query slice (16 dims per lane) + q^2 via 4-lane butterfly
  float4 q[4];
#pragma unroll
  for (int i = 0; i < 4; ++i)
    q[i] = *(const float4*)(query + b * 64 + qs * 16 + i * 4);
  float q2 = 0.f;
#pragma unroll
  for (int i = 0; i < 4; ++i)
    q2 += q[i].x*q[i].x + q[i].y*q[i].y + q[i].z*q[i].z + q[i].w*q[i].w;
  q2 += __shfl_xor(q2, 1);
  q2 += __shfl_xor(q2, 2);

  // LDS byte offsets of this wave's slice in each pipeline stage
  uint32_t lkB[DEPTH], lvB[DEPTH];
#pragma unroll
  for (int i = 0; i < DEPTH; ++i) {
    lkB[i] = (uint32_t)(uintptr_t)&sK[i][wave][0][0];
    lvB[i] = (uint32_t)(uintptr_t)&sV[i][wave][0][0];
  }

  const int segI = lane & 15;   // 16B segment for issue
  const int rI0  = lane >> 4;   // 0..1

  auto issue = [&](int t, int bufSel) {
    const int base = rowBase0 + t * TILE_ROWS + wave * ROWS_PER_WAVE;
    const uint32_t lk = lkB[bufSel];
    const uint32_t lv = lvB[bufSel];
#pragma unroll
    for (int j = 0; j < 4; ++j) {                 // 8 async instrs / wave / tile
      const int r = rI0 + 2 * j;                  // rows 0..7 of the wave slice
      const size_t   go = (size_t)(base + r) * 64 + (size_t)segI * 4;
      const uint32_t lo = (uint32_t)(r * 256 + (((segI + r) & 15) << 4)); // swizzled
      async_b128(lk + lo, keyB + go);
      async_b128(lv + lo, valB + go);
    }
  };

  // online-softmax state (scalars identical across a 4-lane row group)
  float m = -1e30f, s = 0.f, den = 0.f;
  float4 acc[4] = {{0,0,0,0},{0,0,0,0},{0,0,0,0},{0,0,0,0}};

  issue(0, 0);
  if (tiles > 1) issue(1, 1);
  for (int t = 0; t < tiles; ++t) {
    if      (t + 2 < tiles) { issue(t + 2, (t + 2) % DEPTH); wait_async_le16(); }
    else if (t + 1 < tiles) { wait_async_le8(); }
    else                    { wait_async_0();  }

    const int buf = t % DEPTH;
    float4 k4[4], v4[4];
#pragma unroll
    for (int i = 0; i < 4; ++i) {
      const int sw = (((qs * 4 + i) + quad) & 15) * 4;   // un-swizzle
      k4[i] = *(const float4*)&sK[buf][wave][quad][sw];
      v4[i] = *(const float4*)&sV[buf][wave][quad][sw];
    }
    float xy = 0.f, k2 = 0.f, v2 = 0.f;
#pragma unroll
    for (int i = 0; i < 4; ++i) {
      xy += k4[i].x*q[i].x + k4[i].y*q[i].y + k4[i].z*q[i].z + k4[i].w*q[i].w;
      k2 += k4[i].x*k4[i].x + k4[i].y*k4[i].y + k4[i].z*k4[i].z + k4[i].w*k4[i].w;
      v2 += v4[i].x*v4[i].x + v4[i].y*v4[i].y + v4[i].z*v4[i].z + v4[i].w*v4[i].w;
    }
    xy += __shfl_xor(xy, 1);  k2 += __shfl_xor(k2, 1);  v2 += __shfl_xor(v2, 1);
    xy += __shfl_xor(xy, 2);  k2 += __shfl_xor(k2, 2);  v2 += __shfl_xor(v2, 2);

    // mobius_add(-k,q): ||num||^2 = a^2 k2 - 2ab xy + b^2 q2 ; den = 1-2xy+k2*q2
    const float a  = 1.0f - 2.0f * xy + q2;
    const float bc = 1.0f - k2;
    const float dn = fmaxf(1.0f - 2.0f * xy + k2 * q2, 1e-15f);
    const float nd2 = a * a * k2 - 2.0f * a * bc * xy + bc * bc * q2;
    float nd = sqrtf(fmaxf(nd2, 0.0f)) / dn;
    nd = fminf(fmaxf(nd, 0.0f), 1.0f - 1e-5f);
    const float score = beta * __logf((1.0f + nd) / (1.0f - nd)) - cc; // 2*artanh
    const float lamb  = 2.0f / fmaxf(1.0f - v2, 1e-10f);

    const float mOld = m;
    m = fmaxf(m, score);
    const float corr = __expf(mOld - m);
    const float e    = __expf(score - m);
    s   = s   * corr + e;
    den = den * corr + e * (lamb - 1.0f);
    const float w = e * lamb;
#pragma unroll
    for (int i = 0; i < 4; ++i) {
      acc[i].x = acc[i].x * corr + w * v4[i].x;
      acc[i].y = acc[i].y * corr + w * v4[i].y;
      acc[i].z = acc[i].z * corr + w * v4[i].z;
      acc[i].w = acc[i].w * corr + w * v4[i].w;
    }
  }

  // ---- block merge: combine the 32 states that share each 16-dim group ----
  float* st = &sM[tid * 20];
  st[0] = m; st[1] = s; st[2] = den;
#pragma unroll
  for (int i = 0; i < 4; ++i) {
    st[3 + i*4 + 0] = acc[i].x;  st[3 + i*4 + 1] = acc[i].y;
    st[3 + i*4 + 2] = acc[i].z;  st[3 + i*4 + 3] = acc[i].w;
  }
  __syncthreads();

  if (tid < 4) {                 // worker g merges all threads with (tid&3)==g
    float M = -1e30f, S = 0.f, D = 0.f;
    float A[16];
#pragma unroll
    for (int i = 0; i < 16; ++i) A[i] = 0.f;
    for (int j = 0; j < 32; ++j) {
      const float* sj = &sM[(j * 4 + tid) * 20];
      const float Mn = fmaxf(M, sj[0]);
      const float f0 = __expf(M - Mn), f1 = __expf(sj[0] - Mn);
      S = S * f0 + sj[1] * f1;
      D = D * f0 + sj[2] * f1;
#pragma unroll
      for (int i = 0; i < 16; ++i) A[i] = A[i] * f0 + sj[3 + i] * f1;
      M = Mn;
    }
    float* o = partials + (size_t)blockIdx.x * PSTRIDE;
#pragma unroll
    for (int i = 0; i < 4; ++i) {
      float4 v; v.x = A[i*4]; v.y = A[i*4+1]; v.z = A[i*4+2]; v.w = A[i*4+3];
      *(float4*)(o + 4 + tid * 16 + i * 4) = v;
    }
    if (tid == 0) { o[0] = M; o[1] = S; o[2] = D; o[3] = 0.f; }
  }
}

// Pass 2: merge chunk partials per batch, finalize gyro midpoint.
__global__ void __launch_bounds__(64)
hyper_attn_final(const float* __restrict__ partials,
                 float* __restrict__ out, int N)
{
  const int b = blockIdx.x;
  const int d = threadIdx.x;   // 64 dims

  float M = -1e30f, S = 0.f, D = 0.f, Nm = 0.f;
  for (int ch = 0; ch < CHUNKS; ++ch) {
    const float* pp = partials + (size_t)(b * CHUNKS + ch) * PSTRIDE;
    const float mj = pp[0];
    const float Mn = fmaxf(M, mj);
    const float f0 = __expf(M - Mn), f1 = __expf(mj - Mn);
    S  = S  * f0 + pp[1]     * f1;
    D  = D  * f0 + pp[2]     * f1;
    Nm = Nm * f0 + pp[4 + d] * f1;
    M = Mn;
  }
  // (PERTERB terms are > 4 decades below one fp32 ulp of these O(1) values)
  const float denT = D  / S;
  const float numT = Nm / S;
  const float tm   = numT / fmaxf(denT, 1e-10f);     // two_mean[d]

  __shared__ float red[64];
  red[d] = tm * tm;
  __syncthreads();
  if (d == 0) {
    float t = 0.f;
    for (int i = 0; i < 64; ++i) t += red[i];
    red[0] = t;
  }
  __syncthreads();

  const float nrm = sqrtf(red[0]);
  const float nc  = fminf(fmaxf(nrm, 1e-15f), 1.0f - 1e-5f);
  const float ar  = 0.5f * __logf((1.0f + nc) / (1.0f - nc));   // artanh
  const float fac = tanhf(0.5f * ar) / fmaxf(nrm, 1e-15f);      // mobius_scalar_mul(0.5)
  out[b * 64 + d] = (float)N * fac * tm;
}

extern "C" void kernel_launch(void* const* d_in, const int* in_sizes, int n_in,
                              void* d_out, int out_size, void* d_ws, size_t ws_size,
                              hipStream_t stream)
{
  const float* key   = (const float*)d_in[0];
  const float* query = (const float*)d_in[1];
  const float* value = (const float*)d_in[2];
  const float* beta  = (const float*)d_in[3];
  const float* c     = (const float*)d_in[4];

  const int B = in_sizes[1] / 64;            // 16
  const int N = in_sizes[0] / (B * 64);      // 32768

  float* partials = (float*)d_ws;            // B*CHUNKS*PSTRIDE*4 = ~557 KB

  hyper_attn_partial<<<B * CHUNKS, 128, 0, stream>>>(key, query, value, beta, c,
                                                     partials, N);
  hyper_attn_final<<<B, 64, 0, stream>>>(partials, (float*)d_out, N);
}